// GAEConv_24850680775445
// MI455X (gfx1250) — compile-verified
//
#include <hip/hip_runtime.h>
#include <hip/hip_bf16.h>

typedef __attribute__((ext_vector_type(2))) float v2f;
typedef __attribute__((ext_vector_type(8))) float v8f;
typedef __attribute__((ext_vector_type(4))) int   v4i;

#define NN 50000
#define NE 1600000
#define INC 128
#define OC  64
#define NLAYER 4
#define XPAD 132            // X row stride in LDS floats (bank-conflict-free, 16B aligned)
#define WPAD 80             // W swizzled row stride in float2 (half-offset lands on banks 32..63)

#if defined(__has_builtin)
#if __has_builtin(__builtin_amdgcn_global_load_async_to_lds_b128)
#define HAVE_ASYNC_LDS 1
#endif
#if __has_builtin(__builtin_amdgcn_s_wait_asynccnt)
#define HAVE_WAIT_ASYNC 1
#endif
#endif

// ---------------------------------------------------------------- utilities
__global__ void k_fill(float* __restrict__ p, float v, long n) {
    long i = blockIdx.x * (long)blockDim.x + threadIdx.x;
    if (i < n) p[i] = v;
}

__global__ void k_cvt_idx(const long long* __restrict__ ei,
                          int* __restrict__ si, int* __restrict__ di) {
    long e = blockIdx.x * (long)blockDim.x + threadIdx.x;
    if (e >= NE) return;
    si[e] = (int)ei[e];
    di[e] = (int)ei[e + NE];
}

// ---------------------------------------------------------------- GEMM: H = X @ W  (NNx128 @ 128x64)
// Block = 256 threads = 8 waves; wave computes a 16x64 tile with V_WMMA_F32_16X16X4_F32.
// X tile staged in LDS (async global->LDS when available), W staged pre-swizzled so every
// A/B fragment is a single contiguous ds_load_b64. K-loop is branch-free.
__global__ __launch_bounds__(256) void k_gemm(const float* __restrict__ X,
                                              const float* __restrict__ Wg,  // 128x64 (this layer)
                                              float* __restrict__ H) {
    __shared__ float Xl[128 * XPAD];       // 67.6 KB
    __shared__ float Ws[64 * WPAD * 2];    // 40 KB: row r=(k>>1) holds {W[2r][n], W[2r+1][n]}
    const int tid = threadIdx.x;
    const int m0b = blockIdx.x * 128;

    // ---- stage W swizzled: Ws2[r*WPAD + n] = {W[2r][n], W[2r+1][n]},  r = k>>1
    for (int i = tid; i < INC * OC; i += 256) {
        int k = i >> 6, n = i & 63;
        Ws[((k >> 1) * WPAD + n) * 2 + (k & 1)] = Wg[i];
    }

    // ---- stage X tile rows [m0b, m0b+128), zero-fill out-of-range rows
    for (int j = tid; j < 128 * 32; j += 256) {     // 32 x float4 per row
        int row = j >> 5, c4 = (j & 31) * 4;
        int grow = m0b + row;
        float* ldst = &Xl[row * XPAD + c4];
        if (grow < NN) {
#ifdef HAVE_ASYNC_LDS
            __builtin_amdgcn_global_load_async_to_lds_b128(
                (__attribute__((address_space(1))) v4i*)(X + (long)grow * INC + c4),
                (__attribute__((address_space(3))) v4i*)ldst, 0, 0);
#else
            *(float4*)ldst = *(const float4*)(X + (long)grow * INC + c4);
#endif
        } else {
            float4 z; z.x = z.y = z.z = z.w = 0.0f;
            *(float4*)ldst = z;
        }
    }
#ifdef HAVE_ASYNC_LDS
#ifdef HAVE_WAIT_ASYNC
    __builtin_amdgcn_s_wait_asynccnt(0);
#else
    asm volatile("s_wait_asynccnt 0x0" ::: "memory");
#endif
#endif
    __syncthreads();

    const int wave = tid >> 5;
    const int lane = tid & 31;
    const int half = lane >> 4;      // 0: lanes 0-15, 1: lanes 16-31
    const int lm   = lane & 15;
    const int m0   = m0b + wave * 16;

    const v2f* __restrict__ Af = (const v2f*)&Xl[lm * XPAD];  // A row for M=lm
    const v2f* __restrict__ Bf = (const v2f*)Ws;

    v8f c0 = {}, c1 = {}, c2 = {}, c3 = {};

#pragma unroll
    for (int k = 0; k < INC; k += 4) {
        // A fragment (16x4 f32): lane-half h holds K = k+2h, k+2h+1 -> contiguous float2
        v2f a = Af[(k >> 1) + half];
        // B fragments: row (k>>1)+half of swizzled W, col = nt*16 + lm -> contiguous float2
        const v2f* br = &Bf[((k >> 1) + half) * WPAD + lm];
        v2f b0 = br[0];
        v2f b1 = br[16];
        v2f b2 = br[32];
        v2f b3 = br[48];
        c0 = __builtin_amdgcn_wmma_f32_16x16x4_f32(false, a, false, b0, (short)0, c0, false, false);
        c1 = __builtin_amdgcn_wmma_f32_16x16x4_f32(false, a, false, b1, (short)0, c1, false, false);
        c2 = __builtin_amdgcn_wmma_f32_16x16x4_f32(false, a, false, b2, (short)0, c2, false, false);
        c3 = __builtin_amdgcn_wmma_f32_16x16x4_f32(false, a, false, b3, (short)0, c3, false, false);
    }

    // C/D layout: VGPR r -> row = r + 8*half, col = lm
#pragma unroll
    for (int r = 0; r < 8; ++r) {
        int orow = m0 + r + 8 * half;
        if (orow < NN) {
            float* hr = H + (long)orow * OC + lm;
            hr[0]  = c0[r];
            hr[16] = c1[r];
            hr[32] = c2[r];
            hr[48] = c3[r];
        }
    }
}

// ---------------------------------------------------------------- per-node attention scores
__global__ __launch_bounds__(256) void k_scores(const float* __restrict__ H,
                                                const float* __restrict__ asv,
                                                const float* __restrict__ adv,
                                                float* __restrict__ AS,
                                                float* __restrict__ AD) {
    long w   = (blockIdx.x * (long)blockDim.x + threadIdx.x) >> 5;
    int lane = threadIdx.x & 31;
    if (w >= NN) return;
    const float* h = H + w * (long)OC;
    float h0 = h[lane], h1 = h[lane + 32];
    float ps = h0 * asv[lane] + h1 * asv[lane + 32];
    float pd = h0 * adv[lane] + h1 * adv[lane + 32];
    for (int off = 16; off; off >>= 1) {
        ps += __shfl_xor(ps, off, 32);
        pd += __shfl_xor(pd, off, 32);
    }
    if (lane == 0) { AS[w] = ps; AD[w] = pd; }
}

// ---------------------------------------------------------------- edge softmax passes
__device__ __forceinline__ float leaky02(float v) { return v > 0.0f ? v : 0.2f * v; }

__global__ __launch_bounds__(256) void k_edge_max(const int* __restrict__ si,
                                                  const int* __restrict__ di,
                                                  const float* __restrict__ AS,
                                                  const float* __restrict__ AD,
                                                  float* __restrict__ Mx) {
    long e = blockIdx.x * (long)blockDim.x + threadIdx.x;
    if (e >= NE) return;
    int s = si[e], d = di[e];
    float v = leaky02(AS[s] + AD[d]);
    // float atomic max via sign-aware integer atomics (Mx initialized to -1e30)
    if (v >= 0.0f) atomicMax((int*)&Mx[d], __float_as_int(v));
    else           atomicMin((unsigned int*)&Mx[d], __float_as_uint(v));
}

__global__ __launch_bounds__(256) void k_edge_sum(const int* __restrict__ si,
                                                  const int* __restrict__ di,
                                                  const float* __restrict__ AS,
                                                  const float* __restrict__ AD,
                                                  const float* __restrict__ Mx,
                                                  float* __restrict__ SS) {
    long e = blockIdx.x * (long)blockDim.x + threadIdx.x;
    if (e >= NE) return;
    int s = si[e], d = di[e];
    float v = leaky02(AS[s] + AD[d]);
    atomicAdd(&SS[d], expf(v - Mx[d]));
}

// thread per edge: alpha = exp(e - m[d]) / (sum[d]+eps); ATT += alpha
__global__ __launch_bounds__(256) void k_edge_alpha(const int* __restrict__ si,
                                                    const int* __restrict__ di,
                                                    const float* __restrict__ AS,
                                                    const float* __restrict__ AD,
                                                    const float* __restrict__ Mx,
                                                    const float* __restrict__ SS,
                                                    float* __restrict__ ALPHA,
                                                    float* __restrict__ ATT) {
    long e = blockIdx.x * (long)blockDim.x + threadIdx.x;
    if (e >= NE) return;
    int s = si[e], d = di[e];
    float v     = leaky02(AS[s] + AD[d]);
    float alpha = expf(v - Mx[d]) / (SS[d] + 1e-16f);
    ALPHA[e] = alpha;
    ATT[e] += alpha;
}

// wave per edge: OUT[dst] += alpha * H[src]; lane handles channels 2l, 2l+1
__global__ __launch_bounds__(256) void k_edge_scatter(const int* __restrict__ si,
                                                      const int* __restrict__ di,
                                                      const float* __restrict__ ALPHA,
                                                      const float* __restrict__ H,
                                                      float* __restrict__ OUT) {
    long w   = (blockIdx.x * (long)blockDim.x + threadIdx.x) >> 5;
    int lane = threadIdx.x & 31;
    if (w >= NE) return;
    int s = si[w], d = di[w];
    float alpha = ALPHA[w];
    const float2 hv = *(const float2*)(H + (long)s * OC + lane * 2);
    float* o = OUT + (long)d * OC + lane * 2;
    atomicAdd(o,     alpha * hv.x);
    atomicAdd(o + 1, alpha * hv.y);
}

// ---------------------------------------------------------------- per-layer finalize
__global__ __launch_bounds__(256) void k_finalize(const float* __restrict__ OUT,
                                                  const float* __restrict__ bias,
                                                  float* __restrict__ feat) {
    long i = blockIdx.x * (long)blockDim.x + threadIdx.x;
    if (i >= (long)NN * OC) return;
    float v = OUT[i] + bias[i & (OC - 1)];
    v = v > 0.0f ? v : 0.01f * v;
    feat[i] += v;
}

// ---------------------------------------------------------------- pooling over src
__global__ __launch_bounds__(256) void k_pool_edge(const int* __restrict__ si,
                                                   const float* __restrict__ ATT,
                                                   float* __restrict__ CNT,
                                                   float* __restrict__ PS) {
    long e = blockIdx.x * (long)blockDim.x + threadIdx.x;
    if (e >= NE) return;
    int s = si[e];
    atomicAdd(&CNT[s], 1.0f);
    atomicAdd(&PS[s], ATT[e]);
}

__global__ __launch_bounds__(256) void k_pool_final(const float* __restrict__ CNT,
                                                    const float* __restrict__ PS,
                                                    const int* __restrict__ dir,
                                                    float* __restrict__ scores) {
    long n = blockIdx.x * (long)blockDim.x + threadIdx.x;
    if (n >= NN) return;
    float c = CNT[n];
    c = c > 1.0f ? c : 1.0f;
    scores[n] = (float)dir[0] * PS[n] / c;
}

// ---------------------------------------------------------------- driver
extern "C" void kernel_launch(void* const* d_in, const int* in_sizes, int n_in,
                              void* d_out, int out_size, void* d_ws, size_t ws_size,
                              hipStream_t stream) {
    const float*     x       = (const float*)d_in[0];      // [NN,128]
    const float*     W       = (const float*)d_in[1];      // [4,128,64]
    const float*     att_src = (const float*)d_in[2];      // [4,64]
    const float*     att_dst = (const float*)d_in[3];      // [4,64]
    const float*     bias    = (const float*)d_in[4];      // [4,64]
    const long long* ei      = (const long long*)d_in[5];  // [2,NE] int64
    const int*       dir     = (const int*)d_in[6];        // scalar
    float*           out     = (float*)d_out;              // [NN*64] feat ++ [NN] scores

    // workspace layout: ~52 MB, L2-resident
    float* H     = (float*)d_ws;          // NN*64
    float* OUT   = H     + (long)NN * OC; // NN*64
    float* AS    = OUT   + (long)NN * OC; // NN
    float* AD    = AS    + NN;            // NN
    float* Mx    = AD    + NN;            // NN
    float* SS    = Mx    + NN;            // NN
    float* ATT   = SS    + NN;            // NE
    float* ALPHA = ATT   + NE;            // NE
    int*   SRCI  = (int*)(ALPHA + NE);    // NE
    int*   DSTI  = SRCI  + NE;            // NE

    const int T = 256;
    auto blk = [](long n, int t) { return (int)((n + t - 1) / t); };

    // global init
    k_cvt_idx<<<blk(NE, T), T, 0, stream>>>(ei, SRCI, DSTI);
    k_fill<<<blk((long)NN * OC, T), T, 0, stream>>>(out, 0.0f, (long)NN * OC);
    k_fill<<<blk(NE, T), T, 0, stream>>>(ATT, 0.0f, NE);

    for (int i = 0; i < NLAYER; ++i) {
        k_gemm<<<(NN + 127) / 128, T, 0, stream>>>(x, W + (long)i * INC * OC, H);
        k_scores<<<blk((long)NN * 32, T), T, 0, stream>>>(H, att_src + i * OC, att_dst + i * OC, AS, AD);
        k_fill<<<blk(NN, T), T, 0, stream>>>(Mx, -1e30f, NN);
        k_fill<<<blk(NN, T), T, 0, stream>>>(SS, 0.0f, NN);
        k_fill<<<blk((long)NN * OC, T), T, 0, stream>>>(OUT, 0.0f, (long)NN * OC);
        k_edge_max<<<blk(NE, T), T, 0, stream>>>(SRCI, DSTI, AS, AD, Mx);
        k_edge_sum<<<blk(NE, T), T, 0, stream>>>(SRCI, DSTI, AS, AD, Mx, SS);
        k_edge_alpha<<<blk(NE, T), T, 0, stream>>>(SRCI, DSTI, AS, AD, Mx, SS, ALPHA, ATT);
        k_edge_scatter<<<blk((long)NE * 32, T), T, 0, stream>>>(SRCI, DSTI, ALPHA, H, OUT);
        k_finalize<<<blk((long)NN * OC, T), T, 0, stream>>>(OUT, bias + i * OC, out);
    }

    // pooling (reuse Mx as counts, SS as pooled sum)
    k_fill<<<blk(NN, T), T, 0, stream>>>(Mx, 0.0f, NN);
    k_fill<<<blk(NN, T), T, 0, stream>>>(SS, 0.0f, NN);
    k_pool_edge<<<blk(NE, T), T, 0, stream>>>(SRCI, ATT, Mx, SS);
    k_pool_final<<<blk(NN, T), T, 0, stream>>>(Mx, SS, dir, out + (long)NN * OC);
}